// KGEModel_57741540327562
// MI455X (gfx1250) — compile-verified
//
#include <hip/hip_runtime.h>

// Problem constants (from reference)
constexpr int kNE = 200000;
constexpr int kNR = 1000;
constexpr int kNT = 571;
constexpr int kD  = 256;
constexpr int kHD = 2;
constexpr int kHOUSE = 6;
constexpr int kB  = 256;
constexpr int kNEG = 512;
constexpr float kGAMMA = 10.0f;
constexpr float kTHRED = 0.5f;
constexpr float kRTHRED = 0.8f;

// ---- CDNA5 async global->LDS path (gfx1250) --------------------------------
typedef int v4i __attribute__((ext_vector_type(4)));
typedef __attribute__((address_space(1))) v4i gv4i;   // global
typedef __attribute__((address_space(3))) v4i lv4i;   // LDS

#if defined(__gfx1250__) && __has_builtin(__builtin_amdgcn_global_load_async_to_lds_b128)
#define HAVE_ASYNC_LDS 1
#else
#define HAVE_ASYNC_LDS 0
#endif

// Hard-require the async path on the device pass: if this file compiles, the
// gfx1250 async-to-LDS instructions are definitely in the binary.
#if defined(__HIP_DEVICE_COMPILE__) && !HAVE_ASYNC_LDS
#error "gfx1250 async global->LDS builtins not available on this toolchain"
#endif

#if __has_builtin(__builtin_amdgcn_s_wait_asynccnt)
#define WAIT_ASYNC(n) __builtin_amdgcn_s_wait_asynccnt(n)
#elif HAVE_ASYNC_LDS
#define WAIT_ASYNC(n) asm volatile("s_wait_asynccnt " #n ::: "memory")
#else
#define WAIT_ASYNC(n) do {} while (0)
#endif

// ---------------------------------------------------------------------------
// Probe kernel, defined FIRST so the disasm snippet shows the async mnemonic.
// Never launched; exists purely so the compile-only loop can visually confirm
// global_load_async_to_lds_b128 + s_wait_asynccnt in the device assembly.
// ---------------------------------------------------------------------------
__global__ void cdna5_async_probe(const float* __restrict__ in,
                                  float* __restrict__ out) {
  __shared__ __align__(16) float4 buf[32];
#if HAVE_ASYNC_LDS
  __builtin_amdgcn_global_load_async_to_lds_b128(
      (gv4i*)(in + 4 * threadIdx.x),
      (lv4i*)(void*)&buf[threadIdx.x],
      /*offset=*/0, /*cpol=*/0);
  WAIT_ASYNC(0);
#else
  buf[threadIdx.x] = ((const float4*)in)[threadIdx.x];
#endif
  out[threadIdx.x] = buf[threadIdx.x].x + buf[threadIdx.x].w;
}

// Stage one 2KB tail row into LDS as a linear copy.
// Lane L moves 16B chunk idx = L + 32*j, j=0..3 -> 4 coalesced 512B async
// b128 bursts per wave (ASYNCcnt += 4).
__device__ __forceinline__ void stage_tail(const float* __restrict__ gbase,
                                           float2* lbuf, int lane) {
#pragma unroll
  for (int j = 0; j < 4; ++j) {
    const int idx = lane + 32 * j;   // 16-byte chunk index, 0..127
#if HAVE_ASYNC_LDS
    __builtin_amdgcn_global_load_async_to_lds_b128(
        (gv4i*)(gbase + 4 * idx),
        (lv4i*)(void*)((char*)lbuf + 16 * idx),
        /*offset=*/0, /*cpol=*/0);
#else
    ((float4*)lbuf)[idx] = ((const float4*)gbase)[idx];
#endif
  }
}

__global__ __launch_bounds__(256) void kge_house_score(
    const float* __restrict__ ent,      // 0: entity_embedding   (NE, D, HD)
    const float* __restrict__ relemb,   // 1: relation_embedding (NR, D, 12)
    const float* __restrict__ htm,      // 2: head_type_mat      (NT, D, 2)
    const float* __restrict__ ttm,      // 3: tail_type_mat      (NT, D, 2)
    const float* __restrict__ r1dir,    // 4: r1_dir_head        (NT)
    const float* __restrict__ r2dir,    // 5: r2_dir_tail        (NT)
    const float* __restrict__ r1scale,  // 6: r1_scale_head      (NT, D)
    const float* __restrict__ r2scale,  // 7: r2_scale_tail      (NT, D)
    const float* __restrict__ kdh,      // 8: k_dir_head         (NR)
    const float* __restrict__ kdt,      // 9: k_dir_tail         (NR)
    const float* __restrict__ ksh,      // 10: k_scale_head      (NR, D)
    const float* __restrict__ kst,      // 11: k_scale_tail      (NR, D)
    const float* __restrict__ relw,     // 12: relation_weight   (NR, D, HD)
    const int*   __restrict__ htv,      // 13: head_type_vec     (NE)
    const int*   __restrict__ hpart,    // 14: head_part         (B, 3)
    const int*   __restrict__ tpart,    // 15: tail_part         (B, NEG)
    float* __restrict__ out)            // (B, NEG)
{
  __shared__ float2 s_hw[kD];              // rw * fully-reflected head
  __shared__ float2 s_rw[kD];              // relation weight
  __shared__ float2 s_r5[kD];              // normalized rel house-5 vector
  __shared__ float  s_c5[kD];              // 2 + kt
  __shared__ __align__(16) float2 s_stage[8][2][kD];  // per-wave double buffer (32KB)

  const int b   = blockIdx.x;
  const int tid = threadIdx.x;

  const int h_id = hpart[b * 3 + 0];
  const int rel  = hpart[b * 3 + 1];
  const int htyp = htv[h_id];

  // ---------- Phase 0: per-d head pipeline (one thread per d) ----------
  {
    const int d = tid;
    float2 h = ((const float2*)ent)[(size_t)h_id * kD + d];

    // reflect by normalized head-type vector with k = r1_head
    float2 r1 = ((const float2*)htm)[(size_t)htyp * kD + d];
    float inv = 1.0f / fmaxf(sqrtf(r1.x * r1.x + r1.y * r1.y), 1e-12f);
    r1.x *= inv; r1.y *= inv;
    const float k1 = fminf(r1dir[htyp] * fabsf(r1scale[htyp * kD + d]), kRTHRED);
    float f = (2.0f + k1) * (r1.x * h.x + r1.y * h.y);
    h.x -= f * r1.x; h.y -= f * r1.y;

    const float kh = fminf(kdh[rel] * fabsf(ksh[rel * kD + d]), kTHRED);
    const float kt = fminf(kdt[rel] * fabsf(kst[rel * kD + d]), kTHRED);

    const float* rbase = relemb + (size_t)rel * (kD * kHD * kHOUSE) + (size_t)d * (kHD * kHOUSE);
#pragma unroll
    for (int i = 0; i < kHOUSE; ++i) {
      float2 r = { rbase[2 * i], rbase[2 * i + 1] };
      float iv = 1.0f / fmaxf(sqrtf(r.x * r.x + r.y * r.y), 1e-12f);
      r.x *= iv; r.y *= iv;
      if (i == kHOUSE - 1) {            // house 5 -> used on the tail side
        s_r5[d] = r;
        s_c5[d] = 2.0f + kt;
      } else {                          // houses 0..4 reflect the head
        const float cf = (i == 0) ? (2.0f + kh) : 2.0f;
        const float ff = cf * (r.x * h.x + r.y * h.y);
        h.x -= ff * r.x; h.y -= ff * r.y;
      }
    }
    const float2 rw = ((const float2*)relw)[(size_t)rel * kD + d];
    s_rw[d] = rw;
    s_hw[d] = make_float2(rw.x * h.x, rw.y * h.y);
  }
  __syncthreads();

  // ---------- Phase 1: 8 waves x 8 negs, double-buffered async gather ----------
  const int lane = tid & 31;
  const int warp = tid >> 5;
  const int negbase = blockIdx.y * 64 + warp * 8;
  const int* tp = tpart + (size_t)b * kNEG + negbase;

  // Pipeline head: stage neg 0 and resolve its type chain up front.
  int   cur_id   = tp[0];
  stage_tail(ent + (size_t)cur_id * (kD * kHD), s_stage[warp][0], lane);
  int   cur_ttyp = htv[cur_id];
  float cur_rd   = r2dir[cur_ttyp];

  for (int j = 0; j < 8; ++j) {
    int   nxt_id = 0, nxt_ttyp = 0;
    float nxt_rd = 0.0f;
    if (j < 7) {
      nxt_id = tp[j + 1];
      stage_tail(ent + (size_t)nxt_id * (kD * kHD), s_stage[warp][(j + 1) & 1], lane);
      // Overlap next neg's dependent scalar chain with the async wait below.
      nxt_ttyp = htv[nxt_id];
      nxt_rd   = r2dir[nxt_ttyp];
    }

    // Prefetch this neg's L2-resident type-row into registers BEFORE waiting,
    // so no L2 latency sits behind s_wait_asynccnt.
    float2 r2v[8];
    float  scv[8];
#pragma unroll
    for (int jj = 0; jj < 8; ++jj) {
      const int d = lane + 32 * jj;
      r2v[jj] = ((const float2*)ttm)[(size_t)cur_ttyp * kD + d];
      scv[jj] = r2scale[cur_ttyp * kD + d];
    }

    if (j < 7) { WAIT_ASYNC(4); }   // current buffer's 4 bursts done
    else       { WAIT_ASYNC(0); }

    const float2* tb = s_stage[warp][j & 1];
    float acc = 0.0f;
#pragma unroll
    for (int jj = 0; jj < 8; ++jj) {
      const int d = lane + 32 * jj;          // bank-conflict-free b64 LDS reads
      float2 t = tb[d];

      // reflect by normalized tail-type vector with k = r2_tail
      float2 r2 = r2v[jj];
      float iv = 1.0f / fmaxf(sqrtf(r2.x * r2.x + r2.y * r2.y), 1e-12f);
      r2.x *= iv; r2.y *= iv;
      const float c2 = 2.0f + fminf(cur_rd * fabsf(scv[jj]), kRTHRED);
      float f2 = c2 * (r2.x * t.x + r2.y * t.y);
      t.x -= f2 * r2.x; t.y -= f2 * r2.y;

      // reflect by relation house 5 with k = kt (from LDS)
      const float2 r5 = s_r5[d];
      const float f5 = s_c5[d] * (r5.x * t.x + r5.y * t.y);
      t.x -= f5 * r5.x; t.y -= f5 * r5.y;

      const float2 hw = s_hw[d];
      const float2 rw = s_rw[d];
      const float dx = hw.x - rw.x * t.x;
      const float dy = hw.y - rw.y * t.y;
      acc += sqrtf(dx * dx + dy * dy);
    }

    // wave32 reduction over lanes (each lane covered 8 of 256 d's)
#pragma unroll
    for (int off = 16; off >= 1; off >>= 1)
      acc += __shfl_xor(acc, off, 32);

    if (lane == 0)
      out[(size_t)b * kNEG + negbase + j] = kGAMMA - acc;

    cur_id = nxt_id; cur_ttyp = nxt_ttyp; cur_rd = nxt_rd;
  }
}

extern "C" void kernel_launch(void* const* d_in, const int* in_sizes, int n_in,
                              void* d_out, int out_size, void* d_ws, size_t ws_size,
                              hipStream_t stream) {
  (void)in_sizes; (void)n_in; (void)d_ws; (void)ws_size; (void)out_size;
  const float* ent     = (const float*)d_in[0];
  const float* relemb  = (const float*)d_in[1];
  const float* htm     = (const float*)d_in[2];
  const float* ttm     = (const float*)d_in[3];
  const float* r1dir   = (const float*)d_in[4];
  const float* r2dir   = (const float*)d_in[5];
  const float* r1scale = (const float*)d_in[6];
  const float* r2scale = (const float*)d_in[7];
  const float* kdh     = (const float*)d_in[8];
  const float* kdt     = (const float*)d_in[9];
  const float* ksh     = (const float*)d_in[10];
  const float* kst     = (const float*)d_in[11];
  const float* relw    = (const float*)d_in[12];
  const int*   htv     = (const int*)d_in[13];
  const int*   hpart   = (const int*)d_in[14];
  const int*   tpart   = (const int*)d_in[15];
  float* out = (float*)d_out;

  dim3 grid(kB, kNEG / 64);   // (256, 8): one b per block.x, 64 negs per block
  kge_house_score<<<grid, 256, 0, stream>>>(
      ent, relemb, htm, ttm, r1dir, r2dir, r1scale, r2scale,
      kdh, kdt, ksh, kst, relw, htv, hpart, tpart, out);
}